// MinibatchDiscrimination_42898133352734
// MI455X (gfx1250) — compile-verified
//
#include <hip/hip_runtime.h>
#include <hip/hip_bf16.h>

typedef __attribute__((ext_vector_type(2))) float v2f;
typedef __attribute__((ext_vector_type(8))) float v8f;

#define N_ROWS 4096
#define F_DIM  256
#define B_DIM  64
#define C_DIM  8
#define NB     (B_DIM * C_DIM)      // 512 columns of M
#define OUT_STRIDE (F_DIM + B_DIM)  // 320

// ---------------------------------------------------------------------------
// Kernel 1: M[4096,512] = x[4096,256] @ T[256,512] using V_WMMA_F32_16X16X4_F32.
// Each wave produces a 16x32 stripe of M (two 16x16 accumulators sharing A),
// iterating K in steps of 4. 8 waves/block -> 512 blocks cover all tiles.
// ---------------------------------------------------------------------------
__global__ __launch_bounds__(256) void mbd_gemm_wmma(
    const float* __restrict__ x, const float* __restrict__ Tm,
    float* __restrict__ M)
{
    const int lane = threadIdx.x & 31;
    const int wave = threadIdx.x >> 5;
    const int half = lane >> 4;       // 0: K pair {0,1}, 1: K pair {2,3}
    const int l    = lane & 15;

    const int tilesPerRow = NB / 32;             // 16 double-tiles per tile-row
    const int dt   = blockIdx.x * 8 + wave;      // double-tile id (0..4095)
    const int tRow = dt / tilesPerRow;
    const int tCol = dt % tilesPerRow;
    const int row0 = tRow * 16;
    const int col0 = tCol * 32;

    v8f acc0 = {};
    v8f acc1 = {};

    const float* xrow = x + (row0 + l) * F_DIM;

    for (int k = 0; k < F_DIM; k += 4) {
        // A fragment: two consecutive K values for this lane's row
        v2f a = *(const v2f*)(xrow + k + 2 * half);

        const float* tk0 = Tm + (k + 2 * half) * NB;
        const float* tk1 = tk0 + NB;
        v2f b0, b1;
        b0[0] = tk0[col0 + l];
        b0[1] = tk1[col0 + l];
        b1[0] = tk0[col0 + 16 + l];
        b1[1] = tk1[col0 + 16 + l];

        acc0 = __builtin_amdgcn_wmma_f32_16x16x4_f32(
            false, a, false, b0, (short)0, acc0, false, false);
        acc1 = __builtin_amdgcn_wmma_f32_16x16x4_f32(
            false, a, false, b1, (short)0, acc1, false, false);
    }

    // Store D: VGPR r holds row (8*half + r), column l / l+16
    float* mout = M + (row0 + 8 * half) * NB;
#pragma unroll
    for (int r = 0; r < 8; ++r) {
        mout[r * NB + col0 + l]      = acc0[r];
        mout[r * NB + col0 + 16 + l] = acc1[r];
    }
}

// ---------------------------------------------------------------------------
// Kernel 2: out[:, 0:256] = x
// ---------------------------------------------------------------------------
__global__ __launch_bounds__(256) void mbd_copy_x(
    const float* __restrict__ x, float* __restrict__ out)
{
    int idx = blockIdx.x * 256 + threadIdx.x;   // over N*F
    int i = idx >> 8;                           // / 256
    int f = idx & 255;
    out[i * OUT_STRIDE + f] = x[idx];
}

// ---------------------------------------------------------------------------
// Kernel 3: all-pairs L1 distance + exp reduction (VALU/TRANS bound).
//   out[i, 256+b] = sum_j exp(-sum_c |M[i,b,c]-M[j,b,c]|) - 1
// Block = 32 i-rows x 8 b-values (256 threads). j swept in 64-row tiles.
// j-tiles staged with GLOBAL_LOAD_ASYNC_TO_LDS_B128 (ASYNCcnt path, no VGPR
// round-trip), double-buffered so the next tile's DMA overlaps compute.
// ---------------------------------------------------------------------------
#define ITILE  32
#define BCHUNK 8
#define JTILE  64

__global__ __launch_bounds__(256) void mbd_pairs(
    const float* __restrict__ M, float* __restrict__ out)
{
    __shared__ float mj[2][JTILE * 64];         // 2 x 16 KB double buffer

    const int t       = threadIdx.x;
    const int b_local = t & 7;                  // 0..7
    const int i_local = t >> 3;                 // 0..31
    const int i0      = blockIdx.x * ITILE;
    const int bbase   = blockIdx.y * BCHUNK;    // b index base (0..56)
    const int i       = i0 + i_local;

    // This thread's slice of the tile copy: 4 x 16B async transfers
    const int jj_base  = t >> 4;                // q>>4 for p-th pass = p*16 + this
    const int off_base = (t & 15) * 4;

    // Issue one 64-row tile (64 x 64 floats) as async global->LDS copies.
    auto issue_tile = [&](int j0, int buf) {
#pragma unroll
        for (int p = 0; p < 4; ++p) {
            int jj  = p * 16 + jj_base;         // row within tile (0..63)
            int off = off_base;                 // float offset within 64-f row
            unsigned lds_addr =
                (unsigned)(unsigned long long)&mj[buf][jj * 64 + off];
            const float* g = M + (size_t)(j0 + jj) * NB + bbase * C_DIM + off;
            asm volatile("global_load_async_to_lds_b128 %0, %1, off"
                         :: "v"(lds_addr), "v"(g) : "memory");
        }
    };

    // This thread's 8 kernel values M[i, bbase+b_local, 0..7]
    float mi[8];
    {
        const float4* p = (const float4*)(M + i * NB + (bbase + b_local) * C_DIM);
        float4 lo = p[0], hi = p[1];
        mi[0] = lo.x; mi[1] = lo.y; mi[2] = lo.z; mi[3] = lo.w;
        mi[4] = hi.x; mi[5] = hi.y; mi[6] = hi.z; mi[7] = hi.w;
    }

    // Prologue: stage tile 0
    issue_tile(0, 0);
    asm volatile("s_wait_asynccnt 0x0" ::: "memory");
    __syncthreads();

    float acc = 0.0f;
    int buf = 0;

    for (int j0 = 0; j0 < N_ROWS; j0 += JTILE) {
        // Prefetch next tile into the other buffer (DMA overlaps compute)
        if (j0 + JTILE < N_ROWS)
            issue_tile(j0 + JTILE, buf ^ 1);

        const float* tile = &mj[buf][0];
#pragma unroll 4
        for (int jj = 0; jj < JTILE; ++jj) {
            const float* mr = &tile[jj * 64 + b_local * 8];
            float d = 0.0f;
#pragma unroll
            for (int c = 0; c < 8; ++c)
                d += fabsf(mi[c] - mr[c]);      // v_sub + v_add |src|
            acc += __expf(-d);                  // v_exp_f32 (TRANS, coexec)
        }

        // Next tile's async copies complete; all waves sync before swap
        asm volatile("s_wait_asynccnt 0x0" ::: "memory");
        __syncthreads();
        buf ^= 1;
    }

    // subtract self-term exp(0)=1
    out[i * OUT_STRIDE + F_DIM + bbase + b_local] = acc - 1.0f;
}

// ---------------------------------------------------------------------------
extern "C" void kernel_launch(void* const* d_in, const int* in_sizes, int n_in,
                              void* d_out, int out_size, void* d_ws, size_t ws_size,
                              hipStream_t stream)
{
    (void)in_sizes; (void)n_in; (void)out_size; (void)ws_size;

    const float* x  = (const float*)d_in[0];    // [4096, 256]
    const float* Tm = (const float*)d_in[1];    // [256, 64, 8] == [256, 512]
    float*       out = (float*)d_out;           // [4096, 320]
    float*       M   = (float*)d_ws;            // [4096, 512] = 8 MB scratch

    // 4096x512 output tiles -> 4096 double-tiles (16x32) -> 8 per block
    mbd_gemm_wmma<<<512, 256, 0, stream>>>(x, Tm, M);

    // copy x into out[:, :256]
    mbd_copy_x<<<(N_ROWS * F_DIM) / 256, 256, 0, stream>>>(x, out);

    // all-pairs: grid (4096/32, 64/8)
    dim3 grid(N_ROWS / ITILE, B_DIM / BCHUNK);
    mbd_pairs<<<grid, 256, 0, stream>>>(M, out);
}